// RoGERModel_2138893714290
// MI455X (gfx1250) — compile-verified
//
#include <hip/hip_runtime.h>
#include <math.h>

#define NUM_USERS 100000
#define NUM_ITEMS 50000
#define N_NODES   150000
#define NEDGE     500000
#define KDIM      64
#define BATCH_N   16384

typedef float v2f __attribute__((ext_vector_type(2)));
typedef float v8f __attribute__((ext_vector_type(8)));

// ---------------------------------------------------------------- zero fill
__global__ __launch_bounds__(256) void zero_kernel(float* __restrict__ p, int n) {
  int i = blockIdx.x * 256 + threadIdx.x;
  if (i < n) p[i] = 0.0f;
}

// ---------------------------------------------------------------- fused GEMM + edge metric + degree
// One wave (32 lanes) handles a 16-edge tile. Computes pu = F@Wu+b, pi = F@Wi+b
// via V_WMMA_F32_16X16X4_F32 (16 k-steps x 4 n-tiles x 2 matrices = 128 WMMA),
// fuses the three cosine reductions per edge, writes mask[e], mask[e+E] and
// atomically accumulates degree.
__global__ __launch_bounds__(256) void edge_metric_kernel(
    const float* __restrict__ ef,
    const float* __restrict__ Wu, const float* __restrict__ bup,
    const float* __restrict__ Wi, const float* __restrict__ bip,
    const float* __restrict__ x,
    const int* __restrict__ rows, const int* __restrict__ cols,
    const float* __restrict__ L0,
    float* __restrict__ masko, float* __restrict__ deg)
{
  __shared__ float sWu[64 * 64];
  __shared__ float sWi[64 * 64];
  const int t = threadIdx.x;
  {
    const float4* g0 = (const float4*)Wu;
    const float4* g1 = (const float4*)Wi;
    float4* s0 = (float4*)sWu;
    float4* s1 = (float4*)sWi;
    for (int i = t; i < 64 * 64 / 4; i += 256) { s0[i] = g0[i]; s1[i] = g1[i]; }
  }
  __syncthreads();

  const int wave = t >> 5;
  const int lane = t & 31;
  const int hf   = lane >> 4;   // which 16-lane half
  const int l16  = lane & 15;
  const int eb   = (blockIdx.x * 8 + wave) * 16;   // E % 16 == 0 -> tiles fully valid
  if (eb >= NEDGE) return;                         // wave-uniform exit, EXEC stays full

  // A fragments: f32 16x4 layout, lane l covers row M=l%16, K = 4s + 2*hf + {0,1}
  const float* frow = ef + (size_t)(eb + l16) * KDIM;
  v2f af[16];
#pragma unroll
  for (int s = 0; s < 16; ++s) {
    const float* p = frow + 4 * s + 2 * hf;
    af[s][0] = p[0];
    af[s][1] = p[1];
  }

  // Row metadata for the 8 C-rows this lane will see: M = 8*hf + v
  int rv[8], civ[8];
#pragma unroll
  for (int v = 0; v < 8; ++v) {
    int r  = rows[eb + 8 * hf + v];   // user index, in [0, NUM_USERS)
    rv[v]  = r;
    civ[v] = r + NUM_ITEMS;           // ci = r - NUM_USERS + N_NODES
  }

  float s_nu[8], s_au[8], s_bu[8], s_ni[8], s_ai[8], s_bi[8];
#pragma unroll
  for (int v = 0; v < 8; ++v) { s_nu[v]=s_au[v]=s_bu[v]=s_ni[v]=s_ai[v]=s_bi[v]=0.0f; }

  for (int nt = 0; nt < 4; ++nt) {
    const int col = nt * 16 + l16;    // output column this lane holds
    v8f accU = {};
    v8f accI = {};
#pragma unroll
    for (int s = 0; s < 16; ++s) {
      const int k0 = 4 * s + 2 * hf;  // B fragment: rows K=k0,k0+1 at column col
      v2f bw, bwi;
      bw[0]  = sWu[k0 * 64 + col];
      bw[1]  = sWu[(k0 + 1) * 64 + col];
      bwi[0] = sWi[k0 * 64 + col];
      bwi[1] = sWi[(k0 + 1) * 64 + col];
      accU = __builtin_amdgcn_wmma_f32_16x16x4_f32(false, af[s], false, bw,  (short)0, accU, false, false);
      accI = __builtin_amdgcn_wmma_f32_16x16x4_f32(false, af[s], false, bwi, (short)0, accI, false, false);
    }
    const float bU = bup[col];
    const float bI = bip[col];
    // C layout: accU[v] = pu[row 8*hf+v][col]; fold into cosine partial sums
#pragma unroll
    for (int v = 0; v < 8; ++v) {
      float pu = accU[v] + bU;
      float pi = accI[v] + bI;
      float rn = x[(size_t)rv[v]  * KDIM + col];
      float cn = x[(size_t)civ[v] * KDIM + col];
      float pu2 = pu * pu, pi2 = pi * pi;
      float rc = rn * cn, rr2 = rn * rn, cc2 = cn * cn;
      s_nu[v] += rc  * pu2;  s_au[v] += rr2 * pu2;  s_bu[v] += cc2 * pu2;
      s_ni[v] += rc  * pi2;  s_ai[v] += rr2 * pi2;  s_bi[v] += cc2 * pi2;
    }
  }

  // Reduce each row's 6 sums across the 16 lanes of this half (xor stays in half)
#pragma unroll
  for (int off = 1; off < 16; off <<= 1) {
#pragma unroll
    for (int v = 0; v < 8; ++v) {
      s_nu[v] += __shfl_xor(s_nu[v], off);
      s_au[v] += __shfl_xor(s_au[v], off);
      s_bu[v] += __shfl_xor(s_bu[v], off);
      s_ni[v] += __shfl_xor(s_ni[v], off);
      s_ai[v] += __shfl_xor(s_ai[v], off);
      s_bi[v] += __shfl_xor(s_bi[v], off);
    }
  }

  // Lane (16*hf + v), v<8 finalizes edge eb + 8*hf + v using accumulator index v=l16
  if (l16 < 8) {
    float nu=0, au=0, bu2=0, ni=0, ai=0, bi2=0;
#pragma unroll
    for (int v = 0; v < 8; ++v) {
      if (l16 == v) { nu=s_nu[v]; au=s_au[v]; bu2=s_bu[v]; ni=s_ni[v]; ai=s_ai[v]; bi2=s_bi[v]; }
    }
    const int e = eb + 8 * hf + l16;
    float nau = fmaxf(sqrtf(au),  1e-8f);
    float nbu = fmaxf(sqrtf(bu2), 1e-8f);
    float ui  = fmaxf(nu / (nau * nbu), 0.0f);          // relu(cosine)
    float nai = fmaxf(sqrtf(ai),  1e-8f);
    float nbi = fmaxf(sqrtf(bi2), 1e-8f);
    float iu  = fmaxf(ni / (nai * nbi), 0.0f);
    float vA = 0.5f * L0[e]         + 0.5f * ui;        // LM = 0.5
    float vB = 0.5f * L0[e + NEDGE] + 0.5f * iu;
    float mA = (vA >= 0.01f) ? 1.0f : 0.0f;             // EPS = 0.01
    float mB = (vB >= 0.01f) ? 1.0f : 0.0f;
    masko[e]         = mA;
    masko[e + NEDGE] = mB;
    atomicAdd(&deg[rows[e]], mA);                       // rows[e]   = u_e
    atomicAdd(&deg[cols[e]], mB);                       // rows[e+E] = i_e = cols[e]
  }
}

// ---------------------------------------------------------------- dinv
__global__ __launch_bounds__(256) void dinv_kernel(const float* __restrict__ deg,
                                                   float* __restrict__ dinv, int n) {
  int i = blockIdx.x * 256 + threadIdx.x;
  if (i < n) {
    float d = deg[i];
    dinv[i] = (d > 0.0f) ? rsqrtf(fmaxf(d, 1e-12f)) : 0.0f;
  }
}

// ---------------------------------------------------------------- SpMM scatter: x_new[rows[f]] += ew[f]*x[cols[f]]
__global__ __launch_bounds__(256) void spmm_scatter_kernel(
    const int* __restrict__ rows, const int* __restrict__ cols,
    const float* __restrict__ maski, const float* __restrict__ dinv,
    const float* __restrict__ xin, float* __restrict__ xout)
{
  const int wave = threadIdx.x >> 5;
  const int lane = threadIdx.x & 31;
  const int f = blockIdx.x * 8 + wave;        // directed edge id, [0, 2E)
  if (f >= 2 * NEDGE) return;
  const int rr = rows[f];
  const int cc = cols[f];
  const float m = maski[f];
  const float ew = dinv[rr] * m * dinv[cc];
  if (ew == 0.0f) return;                     // wave-uniform skip
  const float2* src = (const float2*)(xin + (size_t)cc * KDIM);
  float2 v = src[lane];
  float* dst = xout + (size_t)rr * KDIM + 2 * lane;
  atomicAdd(dst,     ew * v.x);
  atomicAdd(dst + 1, ew * v.y);
}

// ---------------------------------------------------------------- final scoring
__global__ __launch_bounds__(256) void score_kernel(
    const float* __restrict__ x, const float* __restrict__ Bu,
    const float* __restrict__ Bi, const float* __restrict__ Mu,
    const int* __restrict__ users, const int* __restrict__ items,
    float* __restrict__ out)
{
  const int wave = threadIdx.x >> 5;
  const int lane = threadIdx.x & 31;
  const int b = blockIdx.x * 8 + wave;
  if (b >= BATCH_N) return;
  const int u  = users[b];
  const int it = items[b];
  const float2* gu = (const float2*)(x + (size_t)u * KDIM);
  const float2* gi = (const float2*)(x + (size_t)(NUM_USERS + it) * KDIM);
  float2 a = gu[lane];
  float2 c = gi[lane];
  float d = a.x * c.x + a.y * c.y;
#pragma unroll
  for (int off = 16; off >= 1; off >>= 1) d += __shfl_xor(d, off);
  if (lane == 0) out[b] = d + Bu[u] + Bi[it] + Mu[0];
}

// ---------------------------------------------------------------- host launch
extern "C" void kernel_launch(void* const* d_in, const int* in_sizes, int n_in,
                              void* d_out, int out_size, void* d_ws, size_t ws_size,
                              hipStream_t stream) {
  const float* Gu   = (const float*)d_in[0];
  const float* Gi   = (const float*)d_in[1];
  const float* Bu   = (const float*)d_in[2];
  const float* Bi   = (const float*)d_in[3];
  const float* Mu   = (const float*)d_in[4];
  const float* Wu   = (const float*)d_in[5];
  const float* bu_p = (const float*)d_in[6];
  const float* Wi   = (const float*)d_in[7];
  const float* bi_p = (const float*)d_in[8];
  const float* ef   = (const float*)d_in[9];
  const float* L0   = (const float*)d_in[10];
  const int*   rows = (const int*)d_in[11];
  const int*   cols = (const int*)d_in[12];
  const int*   usr  = (const int*)d_in[13];
  const int*   itm  = (const int*)d_in[14];

  const size_t xElems = (size_t)N_NODES * KDIM;   // 9.6M floats
  char* ws = (char*)d_ws;
  float* xA   = (float*)ws;  ws += xElems * sizeof(float);
  float* xB   = (float*)ws;  ws += xElems * sizeof(float);
  float* mask = (float*)ws;  ws += (size_t)2 * NEDGE * sizeof(float);
  float* deg  = (float*)ws;  ws += (size_t)N_NODES * sizeof(float);
  float* dinv = (float*)ws;

  // x = concat(Gu, Gi)
  hipMemcpyAsync(xA, Gu, (size_t)NUM_USERS * KDIM * sizeof(float),
                 hipMemcpyDeviceToDevice, stream);
  hipMemcpyAsync(xA + (size_t)NUM_USERS * KDIM, Gi,
                 (size_t)NUM_ITEMS * KDIM * sizeof(float),
                 hipMemcpyDeviceToDevice, stream);

  float* xcur = xA;
  float* xnext = xB;
  const int tiles = NEDGE / 16;                 // 31250
  const int edgeBlocks = (tiles + 7) / 8;       // 3907
  const int nodeBlocks = (N_NODES + 255) / 256; // 586
  const int xZeroBlocks = (int)((xElems + 255) / 256);

  for (int layer = 0; layer < 3; ++layer) {
    zero_kernel<<<nodeBlocks, 256, 0, stream>>>(deg, N_NODES);
    edge_metric_kernel<<<edgeBlocks, 256, 0, stream>>>(
        ef, Wu, bu_p, Wi, bi_p, xcur, rows, cols, L0, mask, deg);
    dinv_kernel<<<nodeBlocks, 256, 0, stream>>>(deg, dinv, N_NODES);
    zero_kernel<<<xZeroBlocks, 256, 0, stream>>>(xnext, (int)xElems);
    spmm_scatter_kernel<<<(2 * NEDGE) / 8, 256, 0, stream>>>(
        rows, cols, mask, dinv, xcur, xnext);
    float* tmp = xcur; xcur = xnext; xnext = tmp;
  }

  score_kernel<<<BATCH_N / 8, 256, 0, stream>>>(xcur, Bu, Bi, Mu, usr, itm,
                                                (float*)d_out);
}